// GAT_26792005992482
// MI455X (gfx1250) — compile-verified
//
#include <hip/hip_runtime.h>

// ---------------------------------------------------------------------------
// Relational graph attention (per-relation per-head 32x32 projections, edge
// softmax-less normalization, sparse aggregate, unify GEMM) for MI455X.
//
// Roofline: ~15 GFLOP vs ~600 MB HBM traffic @ 23.3 TB/s -> memory bound,
// so keep fp32 end-to-end and use V_WMMA_F32_16X16X4_F32 for the GEMMs.
// Keys+queries (164 MB) fit in the 192 MB L2, so edge gathers hit L2.
// ---------------------------------------------------------------------------

#define NN   20000      // nodes
#define RR   4          // relations
#define HH   8          // heads
#define SS   32         // head dim
#define EMBD 256        // HH*SS
#define EE   400000     // edges
#define NRR  (NN*RR)    // 80000 rows in the sparse matrix

typedef __attribute__((ext_vector_type(2))) float v2f;
typedef __attribute__((ext_vector_type(8))) float v8f;

// D(16x16,f32) = A(16x4,f32) * B(4x16,f32) + C   -- wave32, EXEC must be ~0
__device__ __forceinline__ v8f wmma4(v2f a, v2f b, v8f c) {
    return __builtin_amdgcn_wmma_f32_16x16x4_f32(
        /*neg_a=*/false, a, /*neg_b=*/false, b,
        /*c_mod=*/(short)0, c, /*reuse_a=*/false, /*reuse_b=*/false);
}

// --------------------------- zero-fill scratch -----------------------------
__global__ __launch_bounds__(256) void k_zero(float* __restrict__ p, int n) {
    int i = blockIdx.x * 256 + threadIdx.x;
    if (i < n) p[i] = 0.0f;
}

// --------------------------- K/Q/V projections -----------------------------
// keys[r,node,h,i] = sum_j tokeys[r,h,i,j] * x[node, h*32+j]   (same for Q,V)
// One wave = one (r,h) x 16-node tile; K=32 -> 8 wmma per 16x16 out tile,
// 2 out tiles (i=0..15,16..31) x 3 weight tensors = 48 wmma / wave.
__global__ __launch_bounds__(256) void k_proj(
    const float* __restrict__ x,
    const float* __restrict__ tokeys,
    const float* __restrict__ toqueries,
    const float* __restrict__ tovals,
    float* __restrict__ keys, float* __restrict__ queries, float* __restrict__ vals)
{
    const int wave = threadIdx.x >> 5;
    const int lane = threadIdx.x & 31;
    const int half = lane >> 4;          // selects K sub-pair / M+8 rows
    const int l    = lane & 15;

    const int t     = blockIdx.x * 8 + wave;       // 0 .. 39999 (exact)
    const int r     = t / (HH * 1250);
    const int rem   = t - r * (HH * 1250);
    const int h     = rem / 1250;
    const int mtile = rem - h * 1250;
    const int node0 = mtile * 16;                  // NN % 16 == 0: full tiles

    const float* xa = x + (size_t)(node0 + l) * EMBD + h * SS;   // A row, lane-private
    const size_t wbase = ((size_t)r * HH + h) * SS * SS;         // W[r][h] is 32x32

    const float* W[3] = { tokeys + wbase, toqueries + wbase, tovals + wbase };
    float*       O[3] = { keys    + (size_t)r * NN * EMBD,
                          queries + (size_t)r * NN * EMBD,
                          vals    + (size_t)r * NN * EMBD };

    for (int m = 0; m < 3; ++m) {
        for (int it = 0; it < 2; ++it) {
            const int i0 = it * 16;
            const float* wrow = W[m] + (size_t)(i0 + l) * SS;    // B col n=l is W row i0+l
            v8f acc = {0.f,0.f,0.f,0.f,0.f,0.f,0.f,0.f};
            #pragma unroll
            for (int kc = 0; kc < 8; ++kc) {
                const int k0 = kc * 4 + half * 2;
                v2f a; a.x = xa[k0];   a.y = xa[k0 + 1];
                v2f b; b.x = wrow[k0]; b.y = wrow[k0 + 1];
                acc = wmma4(a, b, acc);
            }
            #pragma unroll
            for (int v = 0; v < 8; ++v) {
                const int row = node0 + half * 8 + v;            // D layout: M = v (+8)
                O[m][(size_t)row * EMBD + h * SS + i0 + l] = acc[v];
            }
        }
    }
}

// ------------------- per-edge dot products + row sums ----------------------
// dot[h,e] = <keys[pred,sub,h,:], queries[pred,obj,h,:]> ; rowsum += dot
__global__ __launch_bounds__(256) void k_dot(
    const float* __restrict__ keys, const float* __restrict__ queries,
    const int* __restrict__ esub, const int* __restrict__ epred,
    const int* __restrict__ eobj,
    float* __restrict__ dot, float* __restrict__ rowsum)
{
    const int t = blockIdx.x * 256 + threadIdx.x;  // E*H threads, exact
    const int e = t >> 3;
    const int h = t & 7;
    if (e >= EE) return;
    const int p = epred[e], sn = esub[e], on = eobj[e];

    const float4* kp = (const float4*)(keys    + ((size_t)p * NN + sn) * EMBD + h * SS);
    const float4* qp = (const float4*)(queries + ((size_t)p * NN + on) * EMBD + h * SS);
    float acc = 0.f;
    #pragma unroll
    for (int i = 0; i < 8; ++i) {
        float4 a = kp[i], b = qp[i];
        acc += a.x*b.x + a.y*b.y + a.z*b.z + a.w*b.w;
    }
    dot[(size_t)h * EE + e] = acc;
    atomicAdd(&rowsum[(size_t)h * NRR + p * NN + sn], acc);   // global_atomic_add_f32
}

// ------------------------ normalized aggregation ---------------------------
// agg[sub, h*32+s] += (dot/rowsum) * vals[pred,obj,h,s]
// (relation sum folded in here: unify is linear & shared across r, and the
//  reference sums over r after unify, so we can sum before the GEMM)
__global__ __launch_bounds__(256) void k_agg(
    const float* __restrict__ vals, const float* __restrict__ dot,
    const float* __restrict__ rowsum,
    const int* __restrict__ esub, const int* __restrict__ epred,
    const int* __restrict__ eobj,
    float* __restrict__ agg)
{
    const int t = blockIdx.x * 256 + threadIdx.x;
    const int e = t >> 3;
    const int h = t & 7;
    if (e >= EE) return;
    const int p = epred[e], sn = esub[e], on = eobj[e];

    const float att = dot[(size_t)h * EE + e]
                    / rowsum[(size_t)h * NRR + p * NN + sn];
    const float4* vp = (const float4*)(vals + ((size_t)p * NN + on) * EMBD + h * SS);
    float* ap = agg + (size_t)sn * EMBD + h * SS;
    #pragma unroll
    for (int i = 0; i < 8; ++i) {
        float4 v = vp[i];
        atomicAdd(ap + i * 4 + 0, att * v.x);
        atomicAdd(ap + i * 4 + 1, att * v.y);
        atomicAdd(ap + i * 4 + 2, att * v.z);
        atomicAdd(ap + i * 4 + 3, att * v.w);
    }
}

// ------------------------------ unify GEMM ---------------------------------
// out[n,f] = sum_c agg[n,c] * unify_w[f,c]   -> [20000x256]@[256x256]^T
// One wave = one 16x16 tile, K=256 -> 64 chained f32 WMMAs. unify_w (256 KB)
// stays L2/L0 resident across all 1250 M-tiles.
__global__ __launch_bounds__(256) void k_unify(
    const float* __restrict__ agg, const float* __restrict__ uw,
    float* __restrict__ out)
{
    const int wave = threadIdx.x >> 5;
    const int lane = threadIdx.x & 31;
    const int half = lane >> 4;
    const int l    = lane & 15;

    const int t  = blockIdx.x * 8 + wave;   // 0 .. 19999 tiles (exact)
    const int mt = t >> 4;                  // node tile 0..1249
    const int ft = t & 15;                  // f tile 0..15
    const int node0 = mt * 16, f0 = ft * 16;

    const float* arow = agg + (size_t)(node0 + l) * EMBD;
    const float* wrow = uw  + (size_t)(f0 + l) * EMBD;

    v8f acc = {0.f,0.f,0.f,0.f,0.f,0.f,0.f,0.f};
    #pragma unroll 8
    for (int kc = 0; kc < 64; ++kc) {
        const int k0 = kc * 4 + half * 2;
        v2f a; a.x = arow[k0]; a.y = arow[k0 + 1];
        v2f b; b.x = wrow[k0]; b.y = wrow[k0 + 1];
        acc = wmma4(a, b, acc);
    }
    #pragma unroll
    for (int v = 0; v < 8; ++v)
        out[(size_t)(node0 + half * 8 + v) * EMBD + f0 + l] = acc[v];
}

// ---------------------------------------------------------------------------
extern "C" void kernel_launch(void* const* d_in, const int* in_sizes, int n_in,
                              void* d_out, int out_size, void* d_ws, size_t ws_size,
                              hipStream_t stream) {
    (void)in_sizes; (void)n_in; (void)out_size; (void)ws_size;
    const float* x         = (const float*)d_in[0];
    const float* tokeys    = (const float*)d_in[1];
    const float* toqueries = (const float*)d_in[2];
    const float* tovals    = (const float*)d_in[3];
    const float* uw        = (const float*)d_in[4];
    const int*   esub      = (const int*)d_in[5];
    const int*   epred     = (const int*)d_in[6];
    const int*   eobj      = (const int*)d_in[7];
    float*       out       = (float*)d_out;

    // Workspace layout (floats); total = 70,400,000 floats = ~282 MB.
    float* ws      = (float*)d_ws;
    float* keys    = ws;                               // R*N*EMB = 20,480,000
    float* queries = keys    + (size_t)RR * NN * EMBD; // 20,480,000
    float* vals    = queries + (size_t)RR * NN * EMBD; // 20,480,000
    float* dot     = vals    + (size_t)RR * NN * EMBD; // H*E     =  3,200,000
    float* rowsum  = dot     + (size_t)HH * EE;        // H*NR    =    640,000
    float* agg     = rowsum  + (size_t)HH * NRR;       // N*EMB   =  5,120,000

    // rowsum and agg are contiguous -> single zero pass (5,760,000 floats).
    k_zero <<<22500, 256, 0, stream>>>(rowsum, HH * NRR + NN * EMBD);
    k_proj <<<5000,  256, 0, stream>>>(x, tokeys, toqueries, tovals,
                                       keys, queries, vals);
    k_dot  <<<12500, 256, 0, stream>>>(keys, queries, esub, epred, eobj,
                                       dot, rowsum);
    k_agg  <<<12500, 256, 0, stream>>>(vals, dot, rowsum, esub, epred, eobj,
                                       agg);
    k_unify<<<2500,  256, 0, stream>>>(agg, uw, out);
}